// Attention_85658827752062
// MI455X (gfx1250) — compile-verified
//
#include <hip/hip_runtime.h>

// Attention decode (no softmax): B=16, H=32, HD=64, T=8192.
// One workgroup per (b,h); 8 wave32 per workgroup; fused single pass over K/V.

#define B_      16
#define H_      32
#define HD_     64
#define T_      8192
#define NX_     2048
#define WAVES_  8
#define TILES_  (T_ / 16)   // 512 t-tiles of 16

typedef __attribute__((ext_vector_type(2))) float v2f;
typedef __attribute__((ext_vector_type(8))) float v8f;

// SWAPX16: exchange values between lanes L and L^16 (group-of-32 ds_swizzle,
// xor_mask=0x10, and_mask=0x1f -> offset 0x401F).
__device__ __forceinline__ float swap_half(float x) {
    return __int_as_float(__builtin_amdgcn_ds_swizzle(__float_as_int(x), 0x401F));
}

__global__ __launch_bounds__(256) void attn_decode_wmma(
    const float* __restrict__ q, const float* __restrict__ k, const float* __restrict__ v,
    const float* __restrict__ past_k, const float* __restrict__ past_v,
    float* __restrict__ out)
{
    __shared__ float lds_q[HD_];
    __shared__ float lds_kn[HD_];
    __shared__ float lds_vn[HD_];
    __shared__ float lds_part[WAVES_][HD_];

    const int tid  = threadIdx.x;
    const int wave = tid >> 5;
    const int lane = tid & 31;
    const int hi   = lane >> 4;   // which half-wave (selects K-pair 0/1 vs 2/3)
    const int ln   = lane & 15;   // position within half (M or N index)
    const int bh   = blockIdx.x;  // fused (b,h)

    const size_t head_off = (size_t)bh * HD_;               // into q/k/v/out
    const float* Kp = past_k + (size_t)bh * HD_ * T_;       // [64][8192], t contiguous
    const float* Vp = past_v + (size_t)bh * T_  * HD_;      // [8192][64], d contiguous

    if (tid < HD_) {
        lds_q[tid]  = q[head_off + tid];
        lds_kn[tid] = k[head_off + tid];
        lds_vn[tid] = v[head_off + tid];
    }
    __syncthreads();

    // Phase-1 B operand: q broadcast over N, in 16x16x4 B layout.
    // Lane half 'hi' holds K=2*hi, 2*hi+1 for each of the 16 k-steps (d = 4s+k).
    v2f bq[16];
#pragma unroll
    for (int s = 0; s < 16; ++s) {
        const int d0 = 4 * s + 2 * hi;
        bq[s].x = lds_q[d0];
        bq[s].y = lds_q[d0 + 1];
    }

    v8f oacc[4];
#pragma unroll
    for (int g = 0; g < 4; ++g) oacc[g] = v8f{};

    for (int tile = wave; tile < TILES_; tile += WAVES_) {
        const int t0 = tile * 16;

        // ---- Phase 1: w[t0+m] = sum_d q[d] * K[d, t0+m] (16 chained fp32 WMMAs)
        // A[m,k] = K[4s+k, t0+m]: lane half 'hi' loads rows 4s+2hi, 4s+2hi+1.
        v8f wacc = v8f{};
#pragma unroll
        for (int s = 0; s < 16; ++s) {
            const size_t d0 = (size_t)(4 * s + 2 * hi);
            v2f a;
            a.x = __builtin_nontemporal_load(&Kp[ d0      * T_ + t0 + ln]);
            a.y = __builtin_nontemporal_load(&Kp[(d0 + 1) * T_ + t0 + ln]);
            wacc = __builtin_amdgcn_wmma_f32_16x16x4_f32(
                false, a, false, bq[s], (short)0, wacc, false, false);
        }

        // D columns are identical (B was N-broadcast): lanes 0-15 hold w[t0..t0+7]
        // in regs 0..7, lanes 16-31 hold w[t0+8..t0+15]. Swap halves so every
        // lane sees all 16 values.
        float wa[8], wb[8];
#pragma unroll
        for (int r = 0; r < 8; ++r) {
            const float mine  = wacc[r];
            const float other = swap_half(mine);
            wa[r] = hi ? other : mine;   // w[t0 + r]
            wb[r] = hi ? mine  : other;  // w[t0 + 8 + r]
        }

        // ---- Phase 2: out[16g+n] += sum_k w[t0+4s+k] * V[t0+4s+k, 16g+n]
        // A is M-broadcast; B rows of V are contiguous 64B segments.
#pragma unroll
        for (int s = 0; s < 4; ++s) {
            float x0, x1, x2, x3;  // wfull[4s .. 4s+3], constant indices after unroll
            if      (s == 0) { x0 = wa[0]; x1 = wa[1]; x2 = wa[2]; x3 = wa[3]; }
            else if (s == 1) { x0 = wa[4]; x1 = wa[5]; x2 = wa[6]; x3 = wa[7]; }
            else if (s == 2) { x0 = wb[0]; x1 = wb[1]; x2 = wb[2]; x3 = wb[3]; }
            else             { x0 = wb[4]; x1 = wb[5]; x2 = wb[6]; x3 = wb[7]; }
            v2f aw;
            aw.x = hi ? x2 : x0;
            aw.y = hi ? x3 : x1;
            const size_t row = (size_t)t0 + 4 * s + 2 * hi;
#pragma unroll
            for (int g = 0; g < 4; ++g) {
                v2f bv;
                bv.x = __builtin_nontemporal_load(&Vp[ row      * HD_ + g * 16 + ln]);
                bv.y = __builtin_nontemporal_load(&Vp[(row + 1) * HD_ + g * 16 + ln]);
                oacc[g] = __builtin_amdgcn_wmma_f32_16x16x4_f32(
                    false, aw, false, bv, (short)0, oacc[g], false, false);
            }
        }
    }

    // D rows are identical (A was M-broadcast); lanes L and L+16 hold the same
    // column value, so the low half writes this wave's 64 partials.
    if (hi == 0) {
#pragma unroll
        for (int g = 0; g < 4; ++g)
            lds_part[wave][g * 16 + ln] = oacc[g][0];
    }
    __syncthreads();

    // Deterministic fixed-order cross-wave reduction + current-token term.
    if (tid < HD_) {
        float sum = 0.0f;
#pragma unroll
        for (int w = 0; w < WAVES_; ++w) sum += lds_part[w][tid];
        float wT = 0.0f;
#pragma unroll
        for (int d = 0; d < HD_; ++d) wT = fmaf(lds_q[d], lds_kn[d], wT);
        sum = fmaf(wT, lds_vn[tid], sum);
        out[head_off + tid] = sum;
    }
}

extern "C" void kernel_launch(void* const* d_in, const int* in_sizes, int n_in,
                              void* d_out, int out_size, void* d_ws, size_t ws_size,
                              hipStream_t stream) {
    (void)in_sizes; (void)n_in; (void)out_size; (void)d_ws; (void)ws_size;
    const float* q  = (const float*)d_in[0];
    const float* k  = (const float*)d_in[1];
    const float* v  = (const float*)d_in[2];
    const float* pk = (const float*)d_in[3];
    const float* pv = (const float*)d_in[4];
    attn_decode_wmma<<<dim3(B_ * H_), dim3(32 * WAVES_), 0, stream>>>(
        q, k, v, pk, pv, (float*)d_out);
}